// QuantumNATEnhanced_65481071410510
// MI455X (gfx1250) — compile-verified
//
#include <hip/hip_runtime.h>
#include <hip/hip_bf16.h>

typedef __attribute__((ext_vector_type(2))) float v2f;
typedef __attribute__((ext_vector_type(8))) float v8f;

#define TPB 256
#define NLAYERS 3

// ---------------------------------------------------------------------------
// Kernel 0: precompute batch-uniform ansatz trig: cos/sin(params[l,w]*0.5)
// ---------------------------------------------------------------------------
__global__ __launch_bounds__(32) void qnat_prep_kernel(
    const float* __restrict__ params, float* __restrict__ trig)
{
    const int t = threadIdx.x;
    if (t < NLAYERS * 4) {
        const float h = params[t] * 0.5f;
        trig[2 * t + 0] = cosf(h);   // full-precision, runs once
        trig[2 * t + 1] = sinf(h);
    }
}

// ---------------------------------------------------------------------------
// Kernel 1: pool(6x6) -> WMMA f32 16x16x4 projection -> 4-qubit circuit ->
//           unnormalized out + per-block {sum, sumsq} partials
// ---------------------------------------------------------------------------
__global__ __launch_bounds__(TPB) void qnat_main_kernel(
    const float* __restrict__ x, const float* __restrict__ proj_w,
    const float* __restrict__ trig, float* __restrict__ out,
    float* __restrict__ partials)
{
    __shared__ float lds_pool[TPB * 17];   // pooled[sample][16], padded stride
    __shared__ float lds_feats[TPB * 4];   // feats[sample][4]

    const int t = threadIdx.x;
    const int sample = blockIdx.x * TPB + t;

    // ---- 6x6 average pool on 24x24 (1/36 folded into projection B) ----
    const float* xp = x + (size_t)sample * 576;
    float pooled[16];
#pragma unroll
    for (int i = 0; i < 16; ++i) pooled[i] = 0.0f;
#pragma unroll
    for (int r = 0; r < 24; ++r) {
        const int rc = (r / 6) * 4;
#pragma unroll
        for (int c = 0; c < 24; c += 4) {
            float4 v = *reinterpret_cast<const float4*>(xp + r * 24 + c);
            pooled[rc + (c + 0) / 6] += v.x;
            pooled[rc + (c + 1) / 6] += v.y;
            pooled[rc + (c + 2) / 6] += v.z;
            pooled[rc + (c + 3) / 6] += v.w;
        }
    }
#pragma unroll
    for (int i = 0; i < 16; ++i) lds_pool[t * 17 + i] = pooled[i];
    __syncthreads();

    // ---- projection feats(32x4) = pooled(32x16) @ W(16x4) via WMMA f32 ----
    // A 16x4 f32 layout: lanes 0-15: M=0..15, VGPR0=K0, VGPR1=K1;
    //                    lanes 16-31: M=0..15, VGPR0=K2, VGPR1=K3.
    // B 4x16 layout (rows striped across lanes): VGPR0 = rows K0|K2, VGPR1 = K1|K3.
    // D 16x16: lane l -> N = l&15, VGPR r -> M = r + 8*(l>>4).
    {
        const int lane = t & 31;
        const int wavebase = (t >> 5) << 5;     // first local sample of this wave
        const int half = lane >> 4;             // 0 or 1
        const int m = lane & 15;
        const int mm = m & 3;                   // in-bounds proj_w row
        const bool live = (m < 4);
#pragma unroll
        for (int g = 0; g < 2; ++g) {
            const int sb = wavebase + g * 16;   // 16-sample group
            v8f acc = {0.f, 0.f, 0.f, 0.f, 0.f, 0.f, 0.f, 0.f};
#pragma unroll
            for (int k = 0; k < 4; ++k) {
                const int kk = k * 4 + half * 2;
                v2f a, b;
                a.x = lds_pool[(sb + m) * 17 + kk];
                a.y = lds_pool[(sb + m) * 17 + kk + 1];
                const float w0 = proj_w[mm * 16 + kk]     * (1.0f / 36.0f);
                const float w1 = proj_w[mm * 16 + kk + 1] * (1.0f / 36.0f);
                b.x = live ? w0 : 0.0f;         // branchless: EXEC all-1s at WMMA
                b.y = live ? w1 : 0.0f;
                acc = __builtin_amdgcn_wmma_f32_16x16x4_f32(
                    false, a, false, b, (short)0, acc, false, false);
            }
            if (live) {
#pragma unroll
                for (int r = 0; r < 8; ++r)
                    lds_feats[(sb + half * 8 + r) * 4 + m] = acc[r];
            }
        }
    }
    __syncthreads();

    float feats[4];
#pragma unroll
    for (int w = 0; w < 4; ++w) feats[w] = lds_feats[t * 4 + w];

    // ---- 4-qubit statevector in registers; index bit for wire w is (8>>w) ----
    float re[16], im[16];
#pragma unroll
    for (int i = 0; i < 16; ++i) { re[i] = 0.0f; im[i] = 0.0f; }
    re[0] = 1.0f;

    // Encoding: RY(pi*f_w), half-angle h = f * pi/2 ; |h| is small -> v_sin/v_cos
#pragma unroll
    for (int w = 0; w < 4; ++w) {
        const float h = feats[w] * 1.57079632679489662f;
        const float c = __cosf(h), s = __sinf(h);
        const int mask = 8 >> w;
#pragma unroll
        for (int i = 0; i < 16; ++i) {
            if (i & mask) continue;
            const int j = i | mask;
            const float r0 = re[i], i0 = im[i], r1 = re[j], i1 = im[j];
            re[i] = c * r0 - s * r1;  im[i] = c * i0 - s * i1;
            re[j] = s * r0 + c * r1;  im[j] = s * i0 + c * i1;
        }
    }

    // Ansatz: 3 x (RX layer + CNOT chain); trig is batch-uniform -> s_loads
#pragma unroll
    for (int l = 0; l < NLAYERS; ++l) {
#pragma unroll
        for (int w = 0; w < 4; ++w) {
            const float c  = trig[(l * 4 + w) * 2 + 0];
            const float sn = trig[(l * 4 + w) * 2 + 1];
            const int mask = 8 >> w;
#pragma unroll
            for (int i = 0; i < 16; ++i) {
                if (i & mask) continue;
                const int j = i | mask;
                const float r0 = re[i], i0 = im[i], r1 = re[j], i1 = im[j];
                re[i] =  c * r0 + sn * i1;  im[i] =  c * i0 - sn * r1;
                re[j] = sn * i0 +  c * r1;  im[j] = -sn * r0 +  c * i1;
            }
        }
#pragma unroll
        for (int cw = 0; cw < 3; ++cw) {
            const int cmask = 8 >> cw, tmask = 8 >> (cw + 1);
#pragma unroll
            for (int i = 0; i < 16; ++i) {
                if ((i & cmask) && !(i & tmask)) {
                    const int j = i | tmask;
                    float tr = re[i]; re[i] = re[j]; re[j] = tr;
                    float ti = im[i]; im[i] = im[j]; im[j] = ti;
                }
            }
        }
    }

    // PauliZ expvals
    float p[16];
#pragma unroll
    for (int i = 0; i < 16; ++i) p[i] = re[i] * re[i] + im[i] * im[i];
    float out4[4];
#pragma unroll
    for (int w = 0; w < 4; ++w) {
        const int mask = 8 >> w;
        float e = 0.0f;
#pragma unroll
        for (int i = 0; i < 16; ++i) e += (i & mask) ? -p[i] : p[i];
        out4[w] = e;
    }

    *reinterpret_cast<float4*>(out + (size_t)sample * 4) =
        make_float4(out4[0], out4[1], out4[2], out4[3]);

    // ---- deterministic block reduction of {sum, sumsq} per channel ----
#pragma unroll
    for (int j = 0; j < 4; ++j) {
        lds_pool[t * 8 + j]     = out4[j];
        lds_pool[t * 8 + 4 + j] = out4[j] * out4[j];
    }
    __syncthreads();
    for (int s = TPB / 2; s > 0; s >>= 1) {
        if (t < s) {
#pragma unroll
            for (int j = 0; j < 8; ++j)
                lds_pool[t * 8 + j] += lds_pool[(t + s) * 8 + j];
        }
        __syncthreads();
    }
    if (t < 8) partials[blockIdx.x * 8 + t] = lds_pool[t];
}

// ---------------------------------------------------------------------------
// Kernel 2: reduce per-block partials -> per-channel scale/shift (8 floats)
// ---------------------------------------------------------------------------
__global__ __launch_bounds__(256) void qnat_stats_kernel(
    const float* __restrict__ partials, int nblocks,
    const float* __restrict__ gamma, const float* __restrict__ beta,
    float invB, float* __restrict__ stats)
{
    __shared__ float red[256 * 8];
    const int t = threadIdx.x;
    float acc[8];
#pragma unroll
    for (int j = 0; j < 8; ++j) acc[j] = 0.0f;
    for (int b = t; b < nblocks; b += 256)
#pragma unroll
        for (int j = 0; j < 8; ++j) acc[j] += partials[b * 8 + j];
#pragma unroll
    for (int j = 0; j < 8; ++j) red[t * 8 + j] = acc[j];
    __syncthreads();
    for (int s = 128; s > 0; s >>= 1) {
        if (t < s) {
#pragma unroll
            for (int j = 0; j < 8; ++j)
                red[t * 8 + j] += red[(t + s) * 8 + j];
        }
        __syncthreads();
    }
    if (t < 4) {
        const float mean = red[t] * invB;
        const float var  = red[4 + t] * invB - mean * mean;   // biased var
        const float inv  = rsqrtf(var + 1e-5f);
        const float scale = gamma[t] * inv;
        stats[t]     = scale;
        stats[4 + t] = beta[t] - scale * mean;
    }
}

// ---------------------------------------------------------------------------
// Kernel 3: in-place batchnorm apply, one float4 (one sample) per thread
// ---------------------------------------------------------------------------
__global__ __launch_bounds__(256) void qnat_bn_kernel(
    float* __restrict__ out, const float* __restrict__ stats, int nsamples)
{
    const int idx = blockIdx.x * blockDim.x + threadIdx.x;
    if (idx >= nsamples) return;
    const float s0 = stats[0], s1 = stats[1], s2 = stats[2], s3 = stats[3];
    const float h0 = stats[4], h1 = stats[5], h2 = stats[6], h3 = stats[7];
    float4 v = reinterpret_cast<float4*>(out)[idx];
    v.x = v.x * s0 + h0;
    v.y = v.y * s1 + h1;
    v.z = v.z * s2 + h2;
    v.w = v.w * s3 + h3;
    reinterpret_cast<float4*>(out)[idx] = v;
}

// ---------------------------------------------------------------------------
extern "C" void kernel_launch(void* const* d_in, const int* in_sizes, int n_in,
                              void* d_out, int out_size, void* d_ws, size_t ws_size,
                              hipStream_t stream) {
    const float* x      = (const float*)d_in[0];  // (B,1,24,24)
    const float* proj_w = (const float*)d_in[1];  // (4,16)
    const float* params = (const float*)d_in[2];  // (3,4)
    const float* gamma  = (const float*)d_in[3];  // (4,)
    const float* beta   = (const float*)d_in[4];  // (4,)
    float* out = (float*)d_out;                   // (B,4)

    const int B = in_sizes[0] / 576;
    const int nblocks = B / TPB;                  // 256 for B=65536

    float* partials = (float*)d_ws;                    // nblocks*8 floats
    float* stats    = partials + (size_t)nblocks * 8;  // 8 floats
    float* trigbuf  = stats + 8;                       // 24 floats

    qnat_prep_kernel<<<1, 32, 0, stream>>>(params, trigbuf);
    qnat_main_kernel<<<nblocks, TPB, 0, stream>>>(x, proj_w, trigbuf, out, partials);
    qnat_stats_kernel<<<1, 256, 0, stream>>>(partials, nblocks, gamma, beta,
                                             1.0f / (float)B, stats);
    qnat_bn_kernel<<<(B + 255) / 256, 256, 0, stream>>>(out, stats, B);
}